// ClipStyleContrastiveLoss_90013924590031
// MI455X (gfx1250) — compile-verified
//
#include <hip/hip_runtime.h>

#define B_DIM 4096
#define TILE_R 64
#define TILE_C 128
#define LDS_STRIDE 145        // 128 + 17; stride % 64 == 17 (odd) -> conflict-free row & WMMA-B patterns
#define TILES_PER_BLOCK 8     // 512 rows per block
#define THREADS 256           // 8 wave32 waves

typedef float v2f __attribute__((ext_vector_type(2)));
typedef float v8f __attribute__((ext_vector_type(8)));

// ---- CDNA5 async global->LDS copy (ASYNCcnt path), GV addressing form ----
__device__ __forceinline__ void async_load_f32_to_lds(const float* gptr, const float* lds_ptr) {
  unsigned int lds_addr = (unsigned int)(size_t)lds_ptr;       // low 32 bits of generic ptr == LDS offset
  unsigned long long gaddr = (unsigned long long)(size_t)gptr;
  asm volatile("global_load_async_to_lds_b32 %0, %1, off"
               :: "v"(lds_addr), "v"(gaddr) : "memory");
}
__device__ __forceinline__ void wait_asynccnt0() {
  asm volatile("s_wait_asynccnt 0" ::: "memory");
}

// ws layout: [rowS1 | rowS2 | colS1 | colS2], 4*4096 floats (64 KB, L2 resident)
__global__ void hcl_zero_kernel(float* __restrict__ ws, float* __restrict__ out) {
  const int i = (int)(blockIdx.x * blockDim.x + threadIdx.x);
  if (i < 4 * B_DIM) ws[i] = 0.0f;
  if (i == 0) out[0] = 0.0f;
}

__global__ __launch_bounds__(THREADS)
void hcl_sums_kernel(const float* __restrict__ L, float* __restrict__ ws) {
  float* rowS1 = ws;
  float* rowS2 = ws + B_DIM;
  float* colS1 = ws + 2 * B_DIM;
  float* colS2 = ws + 3 * B_DIM;

  __shared__ float tile[TILE_R * LDS_STRIDE];
  __shared__ float rp1[THREADS];
  __shared__ float rp2[THREADS];

  const int t    = (int)threadIdx.x;
  const int lane = t & 31;
  const int wv   = t >> 5;                         // 0..7 -> 16-column group
  const int c0   = (int)blockIdx.x * TILE_C;
  const int rb   = (int)blockIdx.y * (TILE_R * TILES_PER_BLOCK);

  // WMMA column-sum accumulators: persist across all tiles of this block's column strip.
  v8f accC1 = {};   // sum exp(2x) per column
  v8f accC2 = {};   // sum exp(4x) per column
  v2f onesA; onesA[0] = 1.0f; onesA[1] = 1.0f;     // A = all-ones 16x4 (layout-independent)

  for (int ti = 0; ti < TILES_PER_BLOCK; ++ti) {
    const int r0 = rb + ti * TILE_R;

    // ---- stage 64x128 tile into LDS via async tensor path (coalesced 128B/wave) ----
    #pragma unroll
    for (int i = 0; i < 32; ++i) {
      const int linear = i * THREADS + t;          // 0..8191
      const int row = linear >> 7;                 // 0..63
      const int col = linear & (TILE_C - 1);       // 0..127
      async_load_f32_to_lds(L + (size_t)(r0 + row) * B_DIM + (size_t)(c0 + col),
                            &tile[row * LDS_STRIDE + col]);
    }
    wait_asynccnt0();
    __syncthreads();

    // ---- column sums via WMMA: accC += ones(16x4) x E(4x16) => D[m,n] = sum_k E[k,n] + C ----
    {
      const int col  = (wv << 4) + (lane & 15);    // this wave's 16 columns, N = lane%16
      const int rsel = (lane >> 4);                // half-wave row select
      #pragma unroll 4
      for (int k0 = 0; k0 < TILE_R; k0 += 4) {
        const float x0 = tile[(k0 + rsel) * LDS_STRIDE + col];
        const float x1 = tile[(k0 + 2 + rsel) * LDS_STRIDE + col];
        const float e0 = __expf(2.0f * x0);
        const float e1 = __expf(2.0f * x1);
        v2f b1; b1[0] = e0;      b1[1] = e1;
        v2f b2; b2[0] = e0 * e0; b2[1] = e1 * e1;  // exp(4x) = exp(2x)^2
        accC1 = __builtin_amdgcn_wmma_f32_16x16x4_f32(false, onesA, false, b1, (short)0, accC1, false, false);
        accC2 = __builtin_amdgcn_wmma_f32_16x16x4_f32(false, onesA, false, b2, (short)0, accC2, false, false);
      }
    }

    // ---- row sums: thread owns (row, 32-col quarter); stride 145 -> conflict-free ----
    {
      const int rr = t & (TILE_R - 1);
      const int q  = t >> 6;                       // 0..3
      const float* tp = &tile[rr * LDS_STRIDE + q * 32];
      float s1 = 0.0f, s2 = 0.0f;
      #pragma unroll 8
      for (int c = 0; c < 32; ++c) {
        const float e = __expf(2.0f * tp[c]);
        s1 += e;
        s2 += e * e;
      }
      rp1[t] = s1;
      rp2[t] = s2;
    }
    __syncthreads();   // all tile reads + rp writes complete here

    if (t < TILE_R) {  // combine the 4 column-quarters, one atomic pair per row
      const float a1 = rp1[t] + rp1[t + 64] + rp1[t + 128] + rp1[t + 192];
      const float a2 = rp2[t] + rp2[t + 64] + rp2[t + 128] + rp2[t + 192];
      atomicAdd(&rowS1[r0 + t], a1);
      atomicAdd(&rowS2[r0 + t], a2);
    }
    // next iteration's async writes are issued only after the barrier above,
    // and rp rewrites are ordered behind the next post-async __syncthreads.
  }

  // D is uniform over m (ones x B): component 0 of lanes 0..15 holds colsum[lane%16]
  if (lane < 16) {
    const int c = c0 + (wv << 4) + lane;
    atomicAdd(&colS1[c], accC1[0]);
    atomicAdd(&colS2[c], accC2[0]);
  }
}

__global__ __launch_bounds__(THREADS)
void hcl_final_kernel(const float* __restrict__ L, const float* __restrict__ ws,
                      float* __restrict__ out) {
  const float* rowS1 = ws;
  const float* rowS2 = ws + B_DIM;
  const float* colS1 = ws + 2 * B_DIM;
  const float* colS2 = ws + 3 * B_DIM;

  const int i = (int)(blockIdx.x * blockDim.x + threadIdx.x);
  const float d  = L[(size_t)i * B_DIM + i];
  const float p  = __expf(2.0f * d);               // pos = exp(d/t), t=0.5
  const float p2 = p * p;
  const float S1 = rowS1[i] + colS1[i] - 2.0f * p;
  const float S2 = rowS2[i] + colS2[i] - 2.0f * p2;

  const float Nf = 2.0f * (float)B_DIM - 2.0f;     // N = 8190
  const float reweight = Nf * S2 / S1;             // beta=1 => imp==neg
  float Ng = (reweight - 0.1f * Nf * p) * (1.0f / 0.9f);
  Ng = fmaxf(Ng, Nf * 0.13533528323661270f);       // N * exp(-1/t) = N * e^-2
  const float loss = __logf(p + Ng + 1e-8f) - 2.0f * d;

  __shared__ float red[THREADS];
  red[threadIdx.x] = loss * (1.0f / (float)B_DIM); // mean over 2B duplicated rows == mean over B
  __syncthreads();
  for (int s = THREADS / 2; s > 0; s >>= 1) {
    if ((int)threadIdx.x < s) red[threadIdx.x] += red[threadIdx.x + s];
    __syncthreads();
  }
  if (threadIdx.x == 0) atomicAdd(out, red[0]);
}

extern "C" void kernel_launch(void* const* d_in, const int* in_sizes, int n_in,
                              void* d_out, int out_size, void* d_ws, size_t ws_size,
                              hipStream_t stream) {
  (void)in_sizes; (void)n_in; (void)out_size; (void)ws_size;
  const float* logits = (const float*)d_in[0];
  float* ws  = (float*)d_ws;   // needs 4*4096 floats = 64 KB
  float* out = (float*)d_out;

  hcl_zero_kernel<<<(4 * B_DIM + THREADS - 1) / THREADS, THREADS, 0, stream>>>(ws, out);
  hcl_sums_kernel<<<dim3(B_DIM / TILE_C, B_DIM / (TILE_R * TILES_PER_BLOCK)), THREADS, 0, stream>>>(logits, ws);
  hcl_final_kernel<<<B_DIM / THREADS, THREADS, 0, stream>>>(logits, ws, out);
}